// NeualMemoryCell_2362232012941
// MI455X (gfx1250) — compile-verified
//
#include <hip/hip_runtime.h>

// ---------------------------------------------------------------------------
// NeuralMemoryCell for MI455X (gfx1250): five fp32 GEMMs on V_WMMA_F32_16X16X4.
// - TDM (tensor_load_to_lds, TENSORcnt) stages A tiles, double-buffered, with
//   hardware LDS padding for conflict-free banks.
// - B tiles staged pair-interleaved so every WMMA B fragment is one aligned
//   ds_load_b64 (no packing moves), software-pipelined through registers.
// - One barrier per K-chunk; fused activation / grad / learned-optimizer
//   epilogues.
// ---------------------------------------------------------------------------

typedef __attribute__((ext_vector_type(2))) float v2f;
typedef __attribute__((ext_vector_type(4))) float f4;
typedef __attribute__((ext_vector_type(8))) float v8f;
typedef __attribute__((ext_vector_type(4))) unsigned int su4;
typedef __attribute__((ext_vector_type(8))) unsigned int su8;
typedef unsigned int u32;
typedef unsigned long long u64;

#define LRATE 0.01f
#define WDEC  0.01f

#define N_ROWS 16384
#define D_DIM  512
#define H_DIM  1024
#define M_DIM  32

__device__ __forceinline__ v8f wmma_f32(v2f a, v2f b, v8f c) {
  return __builtin_amdgcn_wmma_f32_16x16x4_f32(
      /*neg_a=*/false, a, /*neg_b=*/false, b,
      /*c_mod=*/(short)0, c, /*reuse_a=*/false, /*reuse_b=*/false);
}

// Tensor Data Mover: DMA a 2D fp32 tile [tile_h x tile_w] from a row-major
// global tensor (row stride = stride_w elements) into LDS at lds_off.
// pad=1 adds 4 dwords of LDS padding every 32 dwords (row stride 36 floats).
__device__ __forceinline__ void tdm_load_2d(u32 lds_off,
                                            const float* __restrict__ gptr,
                                            u32 tile_w, u32 tile_h,
                                            u32 tensor_w, u32 tensor_h,
                                            u32 stride_w, u32 pad) {
  u64 ga = (u64)(size_t)gptr;
  su4 g0;
  g0[0] = 1u;                                        // count=1 (valid), user D#
  g0[1] = lds_off;                                   // lds_addr (bytes)
  g0[2] = (u32)ga;                                   // global_addr[31:0]
  g0[3] = ((u32)(ga >> 32) & 0x01FFFFFFu)            // global_addr[56:32]
          | 0x80000000u;                             // type=2 ("image")
  su8 g1;
  // data_size=2 (4B); optional pad_enable(b20), pad_interval=4 (32 dwords,
  // b24:22), pad_amount=3 (4 dwords, b31:25)
  g1[0] = 0x00020000u | (pad ? ((1u << 20) | (4u << 22) | (3u << 25)) : 0u);
  g1[1] = (tensor_w & 0xFFFFu) << 16;                // tensor_dim0[15:0]
  g1[2] = (tensor_w >> 16) | ((tensor_h & 0xFFFFu) << 16);  // dim0 hi, dim1 lo
  g1[3] = (tensor_h >> 16) | (tile_w << 16);         // dim1 hi, tile_dim0
  g1[4] = tile_h & 0xFFFFu;                          // tile_dim1 (tile_dim2=0)
  g1[5] = stride_w;                                  // tensor_dim0_stride[31:0]
  g1[6] = 0u;
  g1[7] = 0u;
  asm volatile("tensor_load_to_lds %0, %1" ::"s"(g0), "s"(g1) : "memory");
}

__device__ __forceinline__ u32 lds_byte_off(const void* p) {
  // Low 32 bits of a flat LDS-aperture address are the LDS byte offset.
  return (u32)(size_t)p;
}

__device__ __forceinline__ float deep_opt(float g,
                                          const float* __restrict__ Wm1,
                                          const float* __restrict__ bm1,
                                          const float* __restrict__ Wm2,
                                          float bm2v) {
  float u = bm2v;
#pragma unroll
  for (int m = 0; m < M_DIM; ++m) u += tanhf(g * Wm1[m] + bm1[m]) * Wm2[m];
  return u;
}

// ---------------------------------------------------------------------------
// Forward GEMM: C = A[rows x K] @ W[K x NC] + bias, 128x128 tile, K-chunk 32.
// A: TDM double-buffered (padded stride 36). B: pair-interleaved LDS, register
// pipelined. One barrier per chunk.
// mode 0: out0 = tanh(C); mode 1: out0 = C (h_t), out1 = (C - xt)*scale (G).
// ---------------------------------------------------------------------------
#define FBM 128
#define FBN 128
#define FKB 32
#define LAS 36    // padded A row stride (floats): conflict-free 64-bank layout
#define LBPS 288  // B row-pair stride (floats): 256 data + 32 pad

__global__ __launch_bounds__(256) void k_fwd(const float* __restrict__ A,
                                             const float* __restrict__ W,
                                             const float* __restrict__ bias,
                                             float* __restrict__ out0,
                                             const float* __restrict__ xt,
                                             float* __restrict__ out1,
                                             int K, int NC, int mode) {
  __shared__ float lA[2][FBM * LAS];
  __shared__ float lB[2][(FKB / 2) * LBPS];
  const int tid  = threadIdx.x;
  const int lane = tid & 31;
  const int wave = tid >> 5;
  const int m0 = blockIdx.x * FBM;
  const int n0 = blockIdx.y * FBN;
  const int wm = (wave & 3) * 32;   // 4 waves along M
  const int wn = (wave >> 2) * 64;  // 2 waves along N

  v8f acc[2][4];
#pragma unroll
  for (int i = 0; i < 2; ++i)
#pragma unroll
    for (int j = 0; j < 4; ++j) acc[i][j] = {};

  const int nIter = K / FKB;
  f4 breg[4];

  // Prologue: stage chunk 0.
  if (wave == 0)
    tdm_load_2d(lds_byte_off(&lA[0][0]), &A[(size_t)m0 * K], FKB, FBM, (u32)K,
                (u32)N_ROWS, (u32)K, 1);
#pragma unroll
  for (int it = 0; it < 4; ++it) {
    int idx = tid + it * 256;
    int r   = idx >> 5;
    int c4  = (idx & 31) << 2;
    f4 v = *(const f4*)&W[(size_t)r * NC + n0 + c4];
#pragma unroll
    for (int e = 0; e < 4; ++e)
      lB[0][(r >> 1) * LBPS + (c4 + e) * 2 + (r & 1)] = v[e];
  }

  for (int itk = 0; itk < nIter; ++itk) {
    const int cur = itk & 1, nxt = cur ^ 1;
    const int kc = itk * FKB;
    if (wave == 0) __builtin_amdgcn_s_wait_tensorcnt(0);  // chunk itk A ready
    __syncthreads();  // all waves done with buf[nxt] reads; buf[cur] visible
    if (itk + 1 < nIter) {
      if (wave == 0)  // overlap next A DMA with this chunk's WMMA loop
        tdm_load_2d(lds_byte_off(&lA[nxt][0]), &A[(size_t)m0 * K + kc + FKB],
                    FKB, FBM, (u32)K, (u32)N_ROWS, (u32)K, 1);
#pragma unroll
      for (int it = 0; it < 4; ++it) {  // issue next B global loads
        int idx = tid + it * 256;
        int r   = idx >> 5;
        int c4  = (idx & 31) << 2;
        breg[it] = *(const f4*)&W[(size_t)(kc + FKB + r) * NC + n0 + c4];
      }
    }
#pragma unroll
    for (int s = 0; s < FKB / 4; ++s) {
      const int kk = s * 4 + ((lane >> 4) << 1);
      const int kp = kk >> 1;
      v2f a[2], b[4];
#pragma unroll
      for (int i = 0; i < 2; ++i) {
        int row = wm + i * 16 + (lane & 15);
        a[i] = *(const v2f*)&lA[cur][row * LAS + kk];
      }
#pragma unroll
      for (int j = 0; j < 4; ++j) {
        int col = wn + j * 16 + (lane & 15);
        b[j] = *(const v2f*)&lB[cur][kp * LBPS + col * 2];
      }
#pragma unroll
      for (int i = 0; i < 2; ++i)
#pragma unroll
        for (int j = 0; j < 4; ++j) acc[i][j] = wmma_f32(a[i], b[j], acc[i][j]);
    }
    if (itk + 1 < nIter) {
#pragma unroll
      for (int it = 0; it < 4; ++it) {
        int idx = tid + it * 256;
        int r   = idx >> 5;
        int c4  = (idx & 31) << 2;
#pragma unroll
        for (int e = 0; e < 4; ++e)
          lB[nxt][(r >> 1) * LBPS + (c4 + e) * 2 + (r & 1)] = breg[it][e];
      }
    }
  }

  const float gscale = 2.0f / ((float)N_ROWS * (float)D_DIM);
#pragma unroll
  for (int i = 0; i < 2; ++i)
#pragma unroll
    for (int j = 0; j < 4; ++j) {
      int colg = n0 + wn + j * 16 + (lane & 15);
      float bb = bias[colg];
#pragma unroll
      for (int e = 0; e < 8; ++e) {
        int rowg = m0 + wm + i * 16 + e + ((lane >> 4) << 3);
        size_t off = (size_t)rowg * NC + colg;
        float v = acc[i][j][e] + bb;
        if (mode == 0) {
          out0[off] = tanhf(v);
        } else {
          out0[off] = v;                          // h_t
          out1[off] = (v - xt[off]) * gscale;     // G
        }
      }
    }
}

// ---------------------------------------------------------------------------
// Weight-grad GEMM + fused learned optimizer:
//   grad[m][n] = sum_k A[k][m] * B[k][n]   (K = 16384)
//   out = P - (LR*deep_opt(grad) + WD*P)
// 64x64 tile, K-chunk 32. A transposed into padded LDS (stride 20),
// B pair-interleaved (stride 160); both register-pipelined, double-buffered,
// one barrier per chunk.
// ---------------------------------------------------------------------------
#define GBM 64
#define GBN 64
#define GKB 32
#define LATS 20   // padded A^T row stride
#define LBGS 160  // B row-pair stride: 128 data + 32 pad

__global__ __launch_bounds__(256) void k_gw(const float* __restrict__ A,
                                            const float* __restrict__ B,
                                            const float* __restrict__ P,
                                            const float* __restrict__ Wm1,
                                            const float* __restrict__ bm1,
                                            const float* __restrict__ Wm2,
                                            const float* __restrict__ bm2,
                                            float* __restrict__ out,
                                            int Ma, int Nb) {
  __shared__ float lAt[2][GBM * LATS];       // [m][k]
  __shared__ float lB[2][(GKB / 2) * LBGS];  // pair-interleaved [k/2][n][2]
  const int tid  = threadIdx.x;
  const int lane = tid & 31;
  const int wave = tid >> 5;
  const int m0 = blockIdx.x * GBM;
  const int n0 = blockIdx.y * GBN;
  const int wm = (wave & 3) * 16;   // 4 waves along M
  const int wn = (wave >> 2) * 32;  // 2 waves along N, 2 tiles each

  v8f acc[2];
  acc[0] = {};
  acc[1] = {};

  const int nIter = N_ROWS / GKB;
  f4 areg[2], breg[2];

  // Prologue: stage chunk 0 (32 x 64 tiles, 2 f4 per thread per matrix).
#pragma unroll
  for (int p = 0; p < 2; ++p) {
    int idx = tid + p * 256;
    int kr  = idx >> 4;
    int c4  = (idx & 15) << 2;
    f4 va = *(const f4*)&A[(size_t)kr * Ma + m0 + c4];
    f4 vb = *(const f4*)&B[(size_t)kr * Nb + n0 + c4];
#pragma unroll
    for (int e = 0; e < 4; ++e) {
      lAt[0][(c4 + e) * LATS + kr] = va[e];
      lB[0][(kr >> 1) * LBGS + (c4 + e) * 2 + (kr & 1)] = vb[e];
    }
  }

  for (int itk = 0; itk < nIter; ++itk) {
    const int cur = itk & 1, nxt = cur ^ 1;
    const int kc = itk * GKB;
    if (itk + 1 < nIter) {  // issue next-chunk global loads
#pragma unroll
      for (int p = 0; p < 2; ++p) {
        int idx = tid + p * 256;
        int kr  = idx >> 4;
        int c4  = (idx & 15) << 2;
        areg[p] = *(const f4*)&A[(size_t)(kc + GKB + kr) * Ma + m0 + c4];
        breg[p] = *(const f4*)&B[(size_t)(kc + GKB + kr) * Nb + n0 + c4];
      }
    }
    __syncthreads();
#pragma unroll
    for (int s = 0; s < GKB / 4; ++s) {
      const int kk = s * 4 + ((lane >> 4) << 1);
      const int kp = kk >> 1;
      int row = wm + (lane & 15);
      v2f a = *(const v2f*)&lAt[cur][row * LATS + kk];
      v2f b[2];
#pragma unroll
      for (int j = 0; j < 2; ++j) {
        int col = wn + j * 16 + (lane & 15);
        b[j] = *(const v2f*)&lB[cur][kp * LBGS + col * 2];
      }
      acc[0] = wmma_f32(a, b[0], acc[0]);
      acc[1] = wmma_f32(a, b[1], acc[1]);
    }
    if (itk + 1 < nIter) {
#pragma unroll
      for (int p = 0; p < 2; ++p) {
        int idx = tid + p * 256;
        int kr  = idx >> 4;
        int c4  = (idx & 15) << 2;
#pragma unroll
        for (int e = 0; e < 4; ++e) {
          lAt[nxt][(c4 + e) * LATS + kr] = areg[p][e];
          lB[nxt][(kr >> 1) * LBGS + (c4 + e) * 2 + (kr & 1)] = breg[p][e];
        }
      }
    }
  }

  const float bm2v = bm2[0];
#pragma unroll
  for (int j = 0; j < 2; ++j) {
    int colg = n0 + wn + j * 16 + (lane & 15);
#pragma unroll
    for (int e = 0; e < 8; ++e) {
      int rowg = m0 + wm + e + ((lane >> 4) << 3);
      size_t off = (size_t)rowg * Nb + colg;
      float g = acc[j][e];
      float u = deep_opt(g, Wm1, bm1, Wm2, bm2v);
      float p = P[off];
      out[off] = p - (LRATE * u + WDEC * p);
    }
  }
}

// ---------------------------------------------------------------------------
// dA = (G @ w2^T) * (1 - Z^2).  G:[N,512] via TDM (padded, double-buffered),
// w2:[1024,512] transposed+pair-interleaved manually, register pipelined.
// ---------------------------------------------------------------------------
__global__ __launch_bounds__(256) void k_dz(const float* __restrict__ G,
                                            const float* __restrict__ W2,
                                            const float* __restrict__ Z,
                                            float* __restrict__ dA) {
  __shared__ float lA[2][FBM * LAS];
  __shared__ float lB[2][(FKB / 2) * LBPS];  // pairs over d: [d/2][h][2]
  const int tid  = threadIdx.x;
  const int lane = tid & 31;
  const int wave = tid >> 5;
  const int m0 = blockIdx.x * FBM;   // over N_ROWS
  const int n0 = blockIdx.y * FBN;   // over H_DIM
  const int wm = (wave & 3) * 32;
  const int wn = (wave >> 2) * 64;

  v8f acc[2][4];
#pragma unroll
  for (int i = 0; i < 2; ++i)
#pragma unroll
    for (int j = 0; j < 4; ++j) acc[i][j] = {};

  const int nIter = D_DIM / FKB;
  f4 breg[4];

  if (wave == 0)
    tdm_load_2d(lds_byte_off(&lA[0][0]), &G[(size_t)m0 * D_DIM], FKB, FBM,
                (u32)D_DIM, (u32)N_ROWS, (u32)D_DIM, 1);
#pragma unroll
  for (int it = 0; it < 4; ++it) {  // w2 tile chunk 0, transposed
    int idx = tid + it * 256;
    int h   = idx >> 3;
    int d4  = (idx & 7) << 2;
    f4 v = *(const f4*)&W2[(size_t)(n0 + h) * D_DIM + d4];
#pragma unroll
    for (int e = 0; e < 4; ++e)
      lB[0][((d4 + e) >> 1) * LBPS + h * 2 + ((d4 + e) & 1)] = v[e];
  }

  for (int itk = 0; itk < nIter; ++itk) {
    const int cur = itk & 1, nxt = cur ^ 1;
    const int kc = itk * FKB;
    if (wave == 0) __builtin_amdgcn_s_wait_tensorcnt(0);
    __syncthreads();
    if (itk + 1 < nIter) {
      if (wave == 0)
        tdm_load_2d(lds_byte_off(&lA[nxt][0]),
                    &G[(size_t)m0 * D_DIM + kc + FKB], FKB, FBM, (u32)D_DIM,
                    (u32)N_ROWS, (u32)D_DIM, 1);
#pragma unroll
      for (int it = 0; it < 4; ++it) {
        int idx = tid + it * 256;
        int h   = idx >> 3;
        int d4  = (idx & 7) << 2;
        breg[it] = *(const f4*)&W2[(size_t)(n0 + h) * D_DIM + kc + FKB + d4];
      }
    }
#pragma unroll
    for (int s = 0; s < FKB / 4; ++s) {
      const int kk = s * 4 + ((lane >> 4) << 1);
      const int kp = kk >> 1;
      v2f a[2], b[4];
#pragma unroll
      for (int i = 0; i < 2; ++i) {
        int row = wm + i * 16 + (lane & 15);
        a[i] = *(const v2f*)&lA[cur][row * LAS + kk];
      }
#pragma unroll
      for (int j = 0; j < 4; ++j) {
        int col = wn + j * 16 + (lane & 15);
        b[j] = *(const v2f*)&lB[cur][kp * LBPS + col * 2];
      }
#pragma unroll
      for (int i = 0; i < 2; ++i)
#pragma unroll
        for (int j = 0; j < 4; ++j) acc[i][j] = wmma_f32(a[i], b[j], acc[i][j]);
    }
    if (itk + 1 < nIter) {
#pragma unroll
      for (int it = 0; it < 4; ++it) {
        int idx = tid + it * 256;
        int h   = idx >> 3;
        int d4  = (idx & 7) << 2;
#pragma unroll
        for (int e = 0; e < 4; ++e)
          lB[nxt][((d4 + e) >> 1) * LBPS + h * 2 + ((d4 + e) & 1)] =
              breg[it][e];
      }
    }
  }

#pragma unroll
  for (int i = 0; i < 2; ++i)
#pragma unroll
    for (int j = 0; j < 4; ++j) {
      int colg = n0 + wn + j * 16 + (lane & 15);
#pragma unroll
      for (int e = 0; e < 8; ++e) {
        int rowg = m0 + wm + i * 16 + e + ((lane >> 4) << 3);
        size_t off = (size_t)rowg * H_DIM + colg;
        float z = Z[off];
        dA[off] = acc[i][j][e] * (1.0f - z * z);
      }
    }
}

// ---------------------------------------------------------------------------
// Bias path: g = colsum(S) over 16384 rows (coalesced), then fused optimizer.
// ---------------------------------------------------------------------------
__global__ __launch_bounds__(256) void k_bias(const float* __restrict__ S,
                                              const float* __restrict__ bP,
                                              const float* __restrict__ Wm1,
                                              const float* __restrict__ bm1,
                                              const float* __restrict__ Wm2,
                                              const float* __restrict__ bm2,
                                              float* __restrict__ out,
                                              int width) {
  int d = blockIdx.x * 256 + threadIdx.x;
  if (d >= width) return;
  float s = 0.f;
  for (int n = 0; n < N_ROWS; ++n) {
    if ((n & 63) == 0 && n + 64 < N_ROWS)
      __builtin_prefetch(&S[(size_t)(n + 64) * width + d], 0, 1);
    s += S[(size_t)n * width + d];
  }
  float u = deep_opt(s, Wm1, bm1, Wm2, bm2[0]);
  float p = bP[d];
  out[d] = p - (LRATE * u + WDEC * p);
}

// ---------------------------------------------------------------------------
extern "C" void kernel_launch(void* const* d_in, const int* in_sizes, int n_in,
                              void* d_out, int out_size, void* d_ws,
                              size_t ws_size, hipStream_t stream) {
  const float* x_t  = (const float*)d_in[0];
  const float* x_tg = (const float*)d_in[1];
  const float* w1   = (const float*)d_in[2];
  const float* b1   = (const float*)d_in[3];
  const float* w2   = (const float*)d_in[4];
  const float* b2   = (const float*)d_in[5];
  const float* Wm1  = (const float*)d_in[6];
  const float* bm1  = (const float*)d_in[7];
  const float* Wm2  = (const float*)d_in[8];
  const float* bm2  = (const float*)d_in[9];

  float* out = (float*)d_out;
  float* ws  = (float*)d_ws;

  float* Z  = ws;                                  // [N, H]
  float* G  = Z + (size_t)N_ROWS * H_DIM;          // [N, D]
  float* dA = G + (size_t)N_ROWS * D_DIM;          // [N, H]

  float* h_out = out;                              // [N, D]
  float* w1o   = h_out + (size_t)N_ROWS * D_DIM;   // [D, H]
  float* b1o   = w1o + (size_t)D_DIM * H_DIM;      // [H]
  float* w2o   = b1o + H_DIM;                      // [H, D]
  float* b2o   = w2o + (size_t)H_DIM * D_DIM;      // [D]

  dim3 blk(256);

  // Z = tanh(x_t @ w1 + b1)
  k_fwd<<<dim3(N_ROWS / FBM, H_DIM / FBN), blk, 0, stream>>>(
      x_t, w1, b1, Z, nullptr, nullptr, D_DIM, H_DIM, 0);
  // h_t = Z @ w2 + b2 ; G = 2*(h_t - x_target)/(N*D)
  k_fwd<<<dim3(N_ROWS / FBM, D_DIM / FBN), blk, 0, stream>>>(
      Z, w2, b2, h_out, x_tg, G, H_DIM, D_DIM, 1);
  // w2' = opt(Z^T @ G)
  k_gw<<<dim3(H_DIM / GBM, D_DIM / GBN), blk, 0, stream>>>(
      Z, G, w2, Wm1, bm1, Wm2, bm2, w2o, H_DIM, D_DIM);
  // b2' = opt(colsum(G))
  k_bias<<<dim3(D_DIM / 256), blk, 0, stream>>>(G, b2, Wm1, bm1, Wm2, bm2, b2o,
                                                D_DIM);
  // dA = (G @ w2^T) * (1 - Z^2)
  k_dz<<<dim3(N_ROWS / FBM, H_DIM / FBN), blk, 0, stream>>>(G, w2, Z, dA);
  // w1' = opt(x_t^T @ dA)
  k_gw<<<dim3(D_DIM / GBM, H_DIM / GBN), blk, 0, stream>>>(
      x_t, dA, w1, Wm1, bm1, Wm2, bm2, w1o, D_DIM, H_DIM);
  // b1' = opt(colsum(dA))
  k_bias<<<dim3(H_DIM / 256), blk, 0, stream>>>(dA, b1, Wm1, bm1, Wm2, bm2, b1o,
                                                H_DIM);
}